// EstimatorQNNHybrid_65481071408594
// MI455X (gfx1250) — compile-verified
//
#include <hip/hip_runtime.h>
#include <math.h>

typedef __attribute__((ext_vector_type(2))) float v2f;
typedef __attribute__((ext_vector_type(8))) float v8f;

namespace {
constexpr int kB    = 4096;
constexpr int kDin  = 64;
constexpr int kDhid = 512;
constexpr int kDlat = 128;
constexpr int kNsup = 512;
constexpr float kGamma = 1.0f;
}

__device__ __forceinline__ v8f wmma_f32_16x16x4(v2f a, v2f b, v8f c) {
  // D = A(16x4 f32) * B(4x16 f32) + C(16x16 f32), wave32
  return __builtin_amdgcn_wmma_f32_16x16x4_f32(false, a, false, b, (short)0, c,
                                               false, false);
}

// Async global->LDS copy, 16 bytes per lane.  dsaddr comes from the low 32
// bits of the generic pointer (LDS aperture: addr[31:0] == wave-relative LDS
// byte offset).  Tracked by ASYNCcnt.
__device__ __forceinline__ void async_ld16(void* lds_dst, const void* gsrc) {
  asm volatile("global_load_async_to_lds_b128 %0, %1, off"
               :
               : "v"((unsigned)(unsigned long long)lds_dst),
                 "v"((unsigned long long)gsrc)
               : "memory");
}
__device__ __forceinline__ void async_wait0() {
  asm volatile("s_wait_asynccnt 0x0" ::: "memory");
}

// ---------------------------------------------------------------------------
// Kernel 1: per support row, cos/sin(s/2) and ||s||^2.  512 blocks x 128 thr.
// ---------------------------------------------------------------------------
__global__ void __launch_bounds__(128)
prep_support_kernel(const float* __restrict__ support,
                    float* __restrict__ cs, float* __restrict__ ss,
                    float* __restrict__ snorm) {
  __shared__ float red[128];
  const int row = blockIdx.x;
  const int t = threadIdx.x;
  const float v = support[row * kDlat + t];
  float s, c;
  sincosf(0.5f * v, &s, &c);
  cs[row * kDlat + t] = c;
  ss[row * kDlat + t] = s;
  red[t] = v * v;
  __syncthreads();
  for (int off = 64; off > 0; off >>= 1) {
    if (t < off) red[t] += red[t + off];
    __syncthreads();
  }
  if (t == 0) snorm[row] = red[0];
}

// ---------------------------------------------------------------------------
// Kernel 2: latent = tanh(X@W1^T + b1)@W2^T + b2 for a 16-row tile.
// Also emits cos(l/2), sin(l/2), ||l||^2.   256 blocks x 128 thr (4 waves).
// ---------------------------------------------------------------------------
__global__ void __launch_bounds__(128)
feature_kernel(const float* __restrict__ x,
               const float* __restrict__ W1, const float* __restrict__ b1,
               const float* __restrict__ W2, const float* __restrict__ b2,
               float* __restrict__ latent, float* __restrict__ cl,
               float* __restrict__ sl, float* __restrict__ lnorm) {
  __shared__ __align__(16) float Xs[16][68];   // stride 272B (x16B, %64==4)
  __shared__ __align__(16) float Hs[16][516];  // stride 2064B (x16B, %64==4)
  __shared__ __align__(16) float Wst[4][16][68];
  __shared__ float lnorm_sm[16];

  const int row0 = blockIdx.x * 16;
  const int tid  = threadIdx.x;
  const int wave = tid >> 5;
  const int lane = tid & 31;
  const int mA   = lane & 15;         // A-row / B-col for WMMA fetch
  const int kh   = (lane >> 4) * 2;   // K sub-offset per lane-half
  const int mbase = (lane < 16) ? 0 : 8;

  // async-stage x tile: 16x64 = 256 float4
  for (int i = tid; i < 16 * 16; i += 128) {
    const int m = i >> 4, c4 = i & 15;
    async_ld16(&Xs[m][c4 * 4], &x[(row0 + m) * kDin + c4 * 4]);
  }
  if (tid < 16) lnorm_sm[tid] = 0.0f;
  async_wait0();
  __syncthreads();

  // ---- GEMM1: H[16,512] = tanh(X @ W1^T + b1), fp32 WMMA, K=64 ----
  for (int j = 0; j < 8; ++j) {
    const int n0 = (wave * 8 + j) * 16;
    for (int i = lane; i < 16 * 16; i += 32) {       // 256 float4
      const int r = i >> 4, c4 = i & 15;
      async_ld16(&Wst[wave][r][c4 * 4], &W1[(n0 + r) * kDin + c4 * 4]);
    }
    async_wait0();
    __syncthreads();
    v8f acc = {};
#pragma unroll
    for (int kk = 0; kk < kDin; kk += 4) {
      v2f a, b;
      a.x = Xs[mA][kk + kh];          a.y = Xs[mA][kk + kh + 1];
      b.x = Wst[wave][mA][kk + kh];   b.y = Wst[wave][mA][kk + kh + 1];
      acc = wmma_f32_16x16x4(a, b, acc);
    }
    const int ncol = n0 + mA;
    const float bn = b1[ncol];
#pragma unroll
    for (int r = 0; r < 8; ++r)
      Hs[mbase + r][ncol] = tanhf(acc[r] + bn);
    __syncthreads();
  }

  // ---- GEMM2: L[16,128] = H @ W2^T + b2, fp32 WMMA, K=512 ----
  for (int t2 = 0; t2 < 2; ++t2) {
    const int n0 = wave * 32 + t2 * 16;
    v8f acc = {};
    for (int kc = 0; kc < kDhid; kc += 64) {
      for (int i = lane; i < 16 * 16; i += 32) {     // 256 float4
        const int r = i >> 4, c4 = i & 15;
        async_ld16(&Wst[wave][r][c4 * 4], &W2[(n0 + r) * kDhid + kc + c4 * 4]);
      }
      async_wait0();
      __syncthreads();
#pragma unroll
      for (int kk = 0; kk < 64; kk += 4) {
        v2f a, b;
        a.x = Hs[mA][kc + kk + kh];     a.y = Hs[mA][kc + kk + kh + 1];
        b.x = Wst[wave][mA][kk + kh];   b.y = Wst[wave][mA][kk + kh + 1];
        acc = wmma_f32_16x16x4(a, b, acc);
      }
      __syncthreads();
    }
    const int ncol = n0 + mA;
    const float bn = b2[ncol];
#pragma unroll
    for (int r = 0; r < 8; ++r) {
      const int m = mbase + r;
      const float l = acc[r] + bn;
      const int g = (row0 + m) * kDlat + ncol;
      latent[g] = l;
      float s, c;
      sincosf(0.5f * l, &s, &c);
      cl[g] = c;
      sl[g] = s;
      atomicAdd(&lnorm_sm[m], l * l);   // ds_add_f32
    }
  }
  __syncthreads();
  if (tid < 16) lnorm[row0 + tid] = lnorm_sm[tid];
}

// ---------------------------------------------------------------------------
// Kernel 3: fused classical RBF (via WMMA dot) + quantum fidelity product
// + weighted reduction.  256 blocks x 128 thr; each wave owns 8 support tiles.
// ---------------------------------------------------------------------------
__global__ void __launch_bounds__(128)
rbf_combine_kernel(const float* __restrict__ latent, const float* __restrict__ cl,
                   const float* __restrict__ sl, const float* __restrict__ lnorm,
                   const float* __restrict__ support, const float* __restrict__ cs,
                   const float* __restrict__ ss, const float* __restrict__ snorm,
                   const float* __restrict__ weights, float* __restrict__ out) {
  __shared__ __align__(16) float Ls[16][132];    // stride 528B (x16B, %64==4)
  __shared__ __align__(16) float CLs[16][132];
  __shared__ __align__(16) float SLs[16][132];
  __shared__ __align__(16) float Sst[4][16][36]; // stride 144B (x16B, %64==36)
  __shared__ __align__(16) float CSst[4][16][36];
  __shared__ __align__(16) float SSst[4][16][36];
  __shared__ __align__(16) float snorm_sm[kNsup];
  __shared__ __align__(16) float wts_sm[kNsup];
  __shared__ float lnorm_sm[16];
  __shared__ float out_sm[16];

  const int row0 = blockIdx.x * 16;
  const int tid  = threadIdx.x;
  const int wave = tid >> 5;
  const int lane = tid & 31;
  const int mA   = lane & 15;
  const int kh   = (lane >> 4) * 2;
  const int mbase = (lane < 16) ? 0 : 8;
  const int nloc = lane & 15;

  // async-stage latent-side tiles: 16x128 floats = 512 float4 per array
  for (int i = tid; i < 16 * 32; i += 128) {
    const int m = i >> 5, c4 = i & 31;
    const int g = (row0 + m) * kDlat + c4 * 4;
    async_ld16(&Ls[m][c4 * 4],  &latent[g]);
    async_ld16(&CLs[m][c4 * 4], &cl[g]);
    async_ld16(&SLs[m][c4 * 4], &sl[g]);
  }
  for (int i = tid; i < kNsup / 4; i += 128) {   // 128 float4 each
    async_ld16(&snorm_sm[i * 4], &snorm[i * 4]);
    async_ld16(&wts_sm[i * 4],   &weights[i * 4]);
  }
  if (tid < 16) {
    lnorm_sm[tid] = lnorm[row0 + tid];
    out_sm[tid]   = 0.0f;
  }
  async_wait0();
  __syncthreads();

  float accO[8];
#pragma unroll
  for (int r = 0; r < 8; ++r) accO[r] = 0.0f;

  for (int j = 0; j < 8; ++j) {
    const int n0 = (wave * 8 + j) * 16;
    v8f dot = {};
    float p[8];
#pragma unroll
    for (int r = 0; r < 8; ++r) p[r] = 1.0f;

    for (int kc = 0; kc < kDlat; kc += 32) {
      // async-stage 16x32 chunk of support / cos(s/2) / sin(s/2)
      for (int i = lane; i < 16 * 8; i += 32) {  // 128 float4 per array
        const int r = i >> 3, c4 = i & 7;
        const int g = (n0 + r) * kDlat + kc + c4 * 4;
        async_ld16(&Sst[wave][r][c4 * 4],  &support[g]);
        async_ld16(&CSst[wave][r][c4 * 4], &cs[g]);
        async_ld16(&SSst[wave][r][c4 * 4], &ss[g]);
      }
      if (j < 7) {  // uniform predicate: warm L2 for the next tile
        __builtin_prefetch(&support[(n0 + 16) * kDlat + kc + lane], 0, 0);
        __builtin_prefetch(&cs[(n0 + 16) * kDlat + kc + lane], 0, 0);
        __builtin_prefetch(&ss[(n0 + 16) * kDlat + kc + lane], 0, 0);
      }
      async_wait0();
      __syncthreads();

      // classical: dot(l, s) via fp32 WMMA
#pragma unroll
      for (int kk = 0; kk < 32; kk += 4) {
        v2f a, b;
        a.x = Ls[mA][kc + kk + kh];     a.y = Ls[mA][kc + kk + kh + 1];
        b.x = Sst[wave][mA][kk + kh];   b.y = Sst[wave][mA][kk + kh + 1];
        dot = wmma_f32_16x16x4(a, b, dot);
      }

      // quantum: prod_i (cos(l/2)cos(s/2) + sin(l/2)sin(s/2))^2 — pure FMA/mul
      for (int i = 0; i < 32; ++i) {
        const float csv = CSst[wave][nloc][i];
        const float ssv = SSst[wave][nloc][i];
#pragma unroll
        for (int r = 0; r < 8; ++r) {
          const float t = CLs[mbase + r][kc + i] * csv +
                          SLs[mbase + r][kc + i] * ssv;
          p[r] *= t * t;
        }
      }
      __syncthreads();
    }

    const float sn = snorm_sm[n0 + nloc];
    const float w  = wts_sm[n0 + nloc];
#pragma unroll
    for (int r = 0; r < 8; ++r) {
      const float sq = lnorm_sm[mbase + r] + sn - 2.0f * dot[r];
      const float kclassical = expf(-kGamma * sq);
      accO[r] += (0.5f * kclassical + 0.5f * p[r]) * w;
    }
  }

#pragma unroll
  for (int r = 0; r < 8; ++r) atomicAdd(&out_sm[mbase + r], accO[r]);
  __syncthreads();
  if (tid < 16) out[row0 + tid] = out_sm[tid];
}

// ---------------------------------------------------------------------------
extern "C" void kernel_launch(void* const* d_in, const int* in_sizes, int n_in,
                              void* d_out, int out_size, void* d_ws, size_t ws_size,
                              hipStream_t stream) {
  (void)in_sizes; (void)n_in; (void)out_size; (void)ws_size;
  const float* x       = (const float*)d_in[0];
  const float* W1      = (const float*)d_in[1];
  const float* b1      = (const float*)d_in[2];
  const float* W2      = (const float*)d_in[3];
  const float* b2      = (const float*)d_in[4];
  const float* support = (const float*)d_in[5];
  const float* weights = (const float*)d_in[6];
  float* out = (float*)d_out;

  float* ws = (float*)d_ws;
  float* latent = ws;                         // 4096*128
  float* cl     = ws + 524288;                // 4096*128
  float* sl     = ws + 1048576;               // 4096*128
  float* lnorm  = ws + 1572864;               // 4096
  float* cs     = ws + 1576960;               // 512*128
  float* ss     = ws + 1642496;               // 512*128
  float* snorm  = ws + 1708032;               // 512

  prep_support_kernel<<<kNsup, 128, 0, stream>>>(support, cs, ss, snorm);
  feature_kernel<<<kB / 16, 128, 0, stream>>>(x, W1, b1, W2, b2,
                                              latent, cl, sl, lnorm);
  rbf_combine_kernel<<<kB / 16, 128, 0, stream>>>(latent, cl, sl, lnorm,
                                                  support, cs, ss, snorm,
                                                  weights, out);
}